// dgl_GCN_43602507989460
// MI455X (gfx1250) — compile-verified
//
#include <hip/hip_runtime.h>
#include <hip/hip_bf16.h>

// GCN on MI455X (gfx1250): fp32 WMMA 16x16x4 for the node GEMMs (fused with the
// D_src^{-1/2} scale), float-atomic scatter over edges, fused bias+ReLU epilogue.
// Scatter traffic (~2.5 GB/pass) dominates; GEMM (~4.1 GFLOP) rides along on WMMA.

typedef __attribute__((ext_vector_type(2))) float v2f;
typedef __attribute__((ext_vector_type(8))) float v8f;

static __host__ __device__ constexpr int ilog2c(int v) {
  return v <= 1 ? 0 : 1 + ilog2c(v >> 1);
}

// ---------------------------------------------------------------- utilities

__global__ void fill_zero_kernel(float* __restrict__ p, long long count) {
  long long i = (long long)blockIdx.x * blockDim.x + threadIdx.x;
  if (i < count) p[i] = 0.0f;
}

__global__ void deg_count_kernel(const int* __restrict__ src,
                                 const int* __restrict__ dst,
                                 float* __restrict__ deg_s,
                                 float* __restrict__ deg_d, int n_edges) {
  int e = blockIdx.x * blockDim.x + threadIdx.x;
  if (e < n_edges) {
    atomicAdd(&deg_s[src[e]], 1.0f);
    atomicAdd(&deg_d[dst[e]], 1.0f);
  }
}

__global__ void inv_sqrt_kernel(float* __restrict__ p, int count) {
  int i = blockIdx.x * blockDim.x + threadIdx.x;
  if (i < count) {
    float d = p[i];
    d = d < 1.0f ? 1.0f : d;       // clip(deg, 1.0)
    p[i] = rsqrtf(d);              // deg^-0.5
  }
}

// ---------------------------------------------------- WMMA GEMM (+ns scale)
// out[n x DOUT] = (A[n x DIN] @ W[DIN x DOUT]) * ns[row]
// One wave32 per 16x16 output tile. fp32 matrix pipe: V_WMMA_F32_16X16X4_F32.
// ISA 7.12.2 layouts (32-bit):
//   A 16x4 : lanes 0-15 -> M=lane, v0=K0 v1=K1 ; lanes 16-31 -> M=lane-16, v0=K2 v1=K3
//   B 4x16 : lanes 0-15 -> N=lane, v0=K0 v1=K1 ; lanes 16-31 -> N=lane-16, v0=K2 v1=K3
//   C 16x16: VGPR r -> M = r + 8*(lane>=16), N = lane&15
template <int DIN, int DOUT>
__global__ void gemm_ns_wmma_kernel(const float* __restrict__ A,
                                    const float* __restrict__ W,
                                    const float* __restrict__ ns,
                                    float* __restrict__ out, int n) {
  constexpr int COL_TILES = (DOUT + 15) / 16;
  int wave = (int)((blockIdx.x * (unsigned)blockDim.x + threadIdx.x) >> 5);
  int lane = threadIdx.x & 31;
  int rowTiles = (n + 15) >> 4;
  if (wave >= rowTiles * COL_TILES) return;   // wave-uniform: EXEC stays all-1s

  int tm = wave / COL_TILES;
  int tn = wave % COL_TILES;
  int laneLo = lane & 15;
  int hi = lane >> 4;                          // 0: K lo pair, 1: K hi pair
  int row = tm * 16 + laneLo;
  int ar = row < n ? row : n - 1;              // clamp (extra rows discarded at store)
  int col = tn * 16 + laneLo;
  const float* __restrict__ arow = A + (long long)ar * DIN;

  v8f acc = {};
#pragma unroll
  for (int k = 0; k < DIN; k += 4) {
    int ka = k + 2 * hi;
    v2f a, b;
    a.x = arow[ka];
    a.y = arow[ka + 1];
    float b0 = 0.0f, b1 = 0.0f;
    if (col < DOUT) {
      b0 = W[(long long)ka * DOUT + col];
      b1 = W[(long long)(ka + 1) * DOUT + col];
    }
    b.x = b0;
    b.y = b1;
    acc = __builtin_amdgcn_wmma_f32_16x16x4_f32(
        /*neg_a=*/false, a, /*neg_b=*/false, b,
        /*c_mod=*/(short)0, acc, /*reuse_a=*/false, /*reuse_b=*/false);
  }

#pragma unroll
  for (int r = 0; r < 8; ++r) {
    int m = tm * 16 + r + 8 * hi;
    int c = tn * 16 + laneLo;
    if (m < n && c < DOUT)
      out[(long long)m * DOUT + c] = acc[r] * ns[m];
  }
}

// -------------------------------------------------------------- edge scatter
// agg[dst[e], f] += tmp[src[e], f]  (one thread per edge x feature; consecutive
// threads cover consecutive features -> coalesced reads and row-local atomics)
template <int DOUT>
__global__ void scatter_add_kernel(const float* __restrict__ tmp,
                                   const int* __restrict__ src,
                                   const int* __restrict__ dst,
                                   float* __restrict__ agg, int n_edges) {
  constexpr int SHIFT = ilog2c(DOUT);
  long long idx = (long long)blockIdx.x * blockDim.x + threadIdx.x;
  long long total = (long long)n_edges << SHIFT;
  if (idx >= total) return;
  int e = (int)(idx >> SHIFT);
  int f = (int)(idx & (DOUT - 1));
  int s = src[e];
  int d = dst[e];
  atomicAdd(&agg[(long long)d * DOUT + f], tmp[(long long)s * DOUT + f]);
}

// ------------------------------------------------------------ bias + ReLU
template <int DOUT>
__global__ void finalize_kernel(const float* __restrict__ agg,
                                const float* __restrict__ nd,
                                const float* __restrict__ bias,
                                float* __restrict__ out, int n) {
  constexpr int SHIFT = ilog2c(DOUT);
  long long idx = (long long)blockIdx.x * blockDim.x + threadIdx.x;
  long long total = (long long)n << SHIFT;
  if (idx >= total) return;
  int m = (int)(idx >> SHIFT);
  int f = (int)(idx & (DOUT - 1));
  float v = fmaf(agg[idx], nd[m], bias[f]);
  out[idx] = v > 0.0f ? v : 0.0f;
}

// ------------------------------------------------------------------- driver

template <int DIN, int DOUT>
static void run_layer(const float* hin, const float* W, const float* bias,
                      const int* src, const int* dst, const float* ns,
                      const float* nd, float* tmp, float* agg, float* hout,
                      int n, int n_edges, hipStream_t stream) {
  constexpr int COL_TILES = (DOUT + 15) / 16;
  int rowTiles = (n + 15) / 16;
  long long tiles = (long long)rowTiles * COL_TILES;
  gemm_ns_wmma_kernel<DIN, DOUT>
      <<<(int)((tiles + 7) / 8), 256, 0, stream>>>(hin, W, ns, tmp, n);

  long long nelem = (long long)n * DOUT;
  fill_zero_kernel<<<(int)((nelem + 255) / 256), 256, 0, stream>>>(agg, nelem);

  long long eelem = (long long)n_edges * DOUT;
  scatter_add_kernel<DOUT>
      <<<(int)((eelem + 255) / 256), 256, 0, stream>>>(tmp, src, dst, agg, n_edges);

  finalize_kernel<DOUT>
      <<<(int)((nelem + 255) / 256), 256, 0, stream>>>(agg, nd, bias, hout, n);
}

extern "C" void kernel_launch(void* const* d_in, const int* in_sizes, int n_in,
                              void* d_out, int out_size, void* d_ws, size_t ws_size,
                              hipStream_t stream) {
  const int* src = (const int*)d_in[0];
  const int* dst = (const int*)d_in[1];
  const float* x = (const float*)d_in[2];
  const float* W1 = (const float*)d_in[3];
  const float* b1 = (const float*)d_in[4];
  const float* W2 = (const float*)d_in[5];
  const float* b2 = (const float*)d_in[6];
  const float* W3 = (const float*)d_in[7];
  const float* b3 = (const float*)d_in[8];
  const float* W4 = (const float*)d_in[9];
  const float* b4 = (const float*)d_in[10];
  const float* W5 = (const float*)d_in[11];
  const float* b5 = (const float*)d_in[12];

  const int n_edges = in_sizes[0];
  const int n = in_sizes[2] / 128;   // 50000 nodes, 128 input features

  // Workspace layout (floats): ns[n] | nd[n] | hbuf[n*128] | tmp[n*128] | agg[n*128]
  float* ws = (float*)d_ws;
  float* ns = ws;
  float* nd = ns + n;
  float* hbuf = nd + n;
  float* tmp = hbuf + (size_t)n * 128;
  float* agg = tmp + (size_t)n * 128;

  // Degrees -> deg^-0.5 (ns, nd are contiguous: transform 2n at once)
  fill_zero_kernel<<<(2 * n + 255) / 256, 256, 0, stream>>>(ns, 2LL * n);
  deg_count_kernel<<<(n_edges + 255) / 256, 256, 0, stream>>>(src, dst, ns, nd, n_edges);
  inv_sqrt_kernel<<<(2 * n + 255) / 256, 256, 0, stream>>>(ns, 2 * n);

  // 5 GraphConv + ReLU layers; last layer writes directly to d_out.
  run_layer<128, 64>(x,    W1, b1, src, dst, ns, nd, tmp, agg, hbuf,         n, n_edges, stream);
  run_layer<64, 128>(hbuf, W2, b2, src, dst, ns, nd, tmp, agg, hbuf,         n, n_edges, stream);
  run_layer<128, 128>(hbuf, W3, b3, src, dst, ns, nd, tmp, agg, hbuf,        n, n_edges, stream);
  run_layer<128, 64>(hbuf, W4, b4, src, dst, ns, nd, tmp, agg, hbuf,         n, n_edges, stream);
  run_layer<64, 1>(hbuf,  W5, b5, src, dst, ns, nd, tmp, agg, (float*)d_out, n, n_edges, stream);
}